// SAGE_37203006718147
// MI455X (gfx1250) — compile-verified
//
#include <hip/hip_runtime.h>

typedef __attribute__((ext_vector_type(2))) float v2f;
typedef __attribute__((ext_vector_type(8))) float v8f;

#define D_FEAT 256
#define ROWS_PER_BLK 64
#define LDS_STRIDE (D_FEAT + 4)   // pad: rows land on distinct banks, keeps 16B align

// ---------------------------------------------------------------------------
// Edge aggregation: agg[dst[e]][:] += h[src[e]][:], deg[dst[e]] += 1.
// One 256-thread block per edge; coalesced 1KB row read; f32 atomics land in
// the L2-resident agg buffer (51MB << 192MB L2).
// ---------------------------------------------------------------------------
__global__ __launch_bounds__(D_FEAT)
void sage_scatter_add(const float* __restrict__ h,
                      const int* __restrict__ src,
                      const int* __restrict__ dst,
                      int E,
                      float* __restrict__ agg,
                      float* __restrict__ deg) {
    int e = blockIdx.x;
    if (e >= E) return;
    int s = src[e];
    int d = dst[e];
    float v = h[(size_t)s * D_FEAT + threadIdx.x];
    atomicAdd(&agg[(size_t)d * D_FEAT + threadIdx.x], v);
    if (threadIdx.x == 0) atomicAdd(&deg[d], 1.0f);
}

// ---------------------------------------------------------------------------
// Fused SAGE GEMM: Out[i,:] = Hself[i,:]@Ws + (Agg[i,:]/max(Deg[i],1))@Wn + b
// Block = 64 output rows x all N cols, 8 waves. Each wave owns 16-col tiles
// (2 for N=256, <=1 for N=64) and register-blocks 4 row tiles, so every B
// fragment feeds 4 WMMAs. N is a template constant so all weight offsets fold
// into global_load immediates; LDS A-fragment bases are precomputed per row
// tile so all ds_load offsets are immediates too.
//
// V_WMMA_F32_16X16X4_F32 VGPR layouts (wave32, ISA 7.12.2):
//   A (16x4):   lanes 0-15 = rows, VGPRv holds K = v + 2*(lane>=16)
//   B (4x16):   lanes 0-15 = cols (mirrored), VGPRv holds K = v + 2*(lane>=16)
//   C/D:        VGPRv -> row v + 8*(lane>=16), col = lane & 15
// ---------------------------------------------------------------------------
template <int N>
__global__ __launch_bounds__(256)
void sage_gemm_wmma(const float* __restrict__ Hself,   // [M x 256]
                    const float* __restrict__ Agg,     // [M x 256]
                    const float* __restrict__ Deg,     // [M]
                    const float* __restrict__ Ws,      // [256 x N]
                    const float* __restrict__ Wn,      // [256 x N]
                    const float* __restrict__ bias,    // [N]
                    float* __restrict__ Out,           // [M x N]
                    int M, int relu) {
    __shared__ float sSelf[ROWS_PER_BLK][LDS_STRIDE];
    __shared__ float sNeigh[ROWS_PER_BLK][LDS_STRIDE];
    __shared__ float sScale[ROWS_PER_BLK];

    const int rowBase = blockIdx.x * ROWS_PER_BLK;
    const int tid  = threadIdx.x;
    const int lane = tid & 31;
    const int wave = tid >> 5;

    // Per-row 1/max(deg,1)
    if (tid < ROWS_PER_BLK) {
        int grow = rowBase + tid;
        float dg = (grow < M) ? Deg[grow] : 1.0f;
        sScale[tid] = 1.0f / fmaxf(dg, 1.0f);
    }
    __syncthreads();

    // Stage self + normalized-neighbor tiles (64 x 256 each) via float4.
    {
        const int c4 = tid & 63;          // float4 column slot
        const int r0 = tid >> 6;          // starting row (0..3)
        #pragma unroll
        for (int rr = 0; rr < ROWS_PER_BLK / 4; ++rr) {
            int r = r0 + rr * 4;
            int grow = rowBase + r;
            float4 sv = make_float4(0.f, 0.f, 0.f, 0.f);
            float4 nv = sv;
            if (grow < M) {
                sv = reinterpret_cast<const float4*>(Hself + (size_t)grow * D_FEAT)[c4];
                nv = reinterpret_cast<const float4*>(Agg  + (size_t)grow * D_FEAT)[c4];
                float sc = sScale[r];
                nv.x *= sc; nv.y *= sc; nv.z *= sc; nv.w *= sc;
            }
            *reinterpret_cast<float4*>(&sSelf[r][c4 * 4])  = sv;
            *reinterpret_cast<float4*>(&sNeigh[r][c4 * 4]) = nv;
        }
    }
    __syncthreads();

    const int row   = lane & 15;   // A-matrix row within a 16-row tile
    const int kh    = lane >> 4;   // K half-select (0: K=v+0, 1: K=v+2)
    const int col16 = lane & 15;   // B/D column within tile

    // Precompute per-row-tile LDS fragment base pointers (lane-local).
    // Fragment for k-step k0 lives at pS[rt] + k0 (8B-aligned b64 pair);
    // within an unroll window k0*4 bytes fits the ds_load immediate field.
    const float* pS[4];
    const float* pN[4];
    #pragma unroll
    for (int rt = 0; rt < 4; ++rt) {
        pS[rt] = &sSelf[rt * 16 + row][2 * kh];
        pN[rt] = &sNeigh[rt * 16 + row][2 * kh];
    }

    // Col tiles: wave, wave+8, ... (N=256 -> 2 per wave, N=64 -> waves 0-3 only)
    for (int ct = wave; ct * 16 < N; ct += 8) {
        const int colBase = ct * 16;
        // One base address per wave; all k-step offsets are immediates.
        const float* wsBase = Ws + 2 * kh * N + colBase + col16;
        const float* wnBase = Wn + 2 * kh * N + colBase + col16;

        v8f acc[4] = {v8f{}, v8f{}, v8f{}, v8f{}};

        #pragma unroll 8
        for (int k0 = 0; k0 < D_FEAT; k0 += 4) {
            v2f bs, bn;
            bs.x = wsBase[k0 * N];
            bs.y = wsBase[(k0 + 1) * N];
            bn.x = wnBase[k0 * N];
            bn.y = wnBase[(k0 + 1) * N];

            #pragma unroll
            for (int rt = 0; rt < 4; ++rt) {
                v2f as, an;
                as.x = pS[rt][k0];
                as.y = pS[rt][k0 + 1];
                an.x = pN[rt][k0];
                an.y = pN[rt][k0 + 1];
                acc[rt] = __builtin_amdgcn_wmma_f32_16x16x4_f32(
                    false, as, false, bs, (short)0, acc[rt], false, false);
                acc[rt] = __builtin_amdgcn_wmma_f32_16x16x4_f32(
                    false, an, false, bn, (short)0, acc[rt], false, false);
            }
        }

        const int ocol = colBase + col16;
        const float b = bias[ocol];
        #pragma unroll
        for (int rt = 0; rt < 4; ++rt) {
            #pragma unroll
            for (int v = 0; v < 8; ++v) {
                int orow = rowBase + rt * 16 + v + 8 * kh;
                if (orow < M) {
                    float val = acc[rt][v] + b;
                    if (relu) val = fmaxf(val, 0.0f);
                    Out[(size_t)orow * N + ocol] = val;
                }
            }
        }
    }
}

// ---------------------------------------------------------------------------
// Host-side orchestration (graph-capture safe: only async ops on `stream`).
// ---------------------------------------------------------------------------
extern "C" void kernel_launch(void* const* d_in, const int* in_sizes, int n_in,
                              void* d_out, int out_size, void* d_ws, size_t ws_size,
                              hipStream_t stream) {
    (void)in_sizes; (void)n_in; (void)out_size; (void)ws_size;

    const int N1 = 50000, N2 = 12000, N3 = 3000;
    const int E0 = 500000, E1 = 120000, E2 = 30000;
    const int DH = 256;

    const float* x    = (const float*)d_in[0];
    const int*   src0 = (const int*)d_in[1];
    const int*   dst0 = (const int*)d_in[2];
    const int*   src1 = (const int*)d_in[3];
    const int*   dst1 = (const int*)d_in[4];
    const int*   src2 = (const int*)d_in[5];
    const int*   dst2 = (const int*)d_in[6];
    // d_in[7..9] = num_dst scalars (compile-time constants here)
    const float* Ws0 = (const float*)d_in[10];
    const float* Wn0 = (const float*)d_in[11];
    const float* b0  = (const float*)d_in[12];
    const float* Ws1 = (const float*)d_in[13];
    const float* Wn1 = (const float*)d_in[14];
    const float* b1  = (const float*)d_in[15];
    const float* Ws2 = (const float*)d_in[16];
    const float* Wn2 = (const float*)d_in[17];
    const float* b2  = (const float*)d_in[18];

    // Workspace (floats): agg (N1*256) | deg (N1) | h1 (N1*256) | h2 (N2*256)
    // deg packed right after each layer's live agg region -> single memset.
    float* agg = (float*)d_ws;
    float* h1  = agg + (size_t)N1 * DH + N1;
    float* h2  = h1 + (size_t)N1 * DH;
    float* out = (float*)d_out;

    // ---- Layer 0: x -> h1 (N1 x 256) ----
    {
        float* deg = agg + (size_t)N1 * DH;
        hipMemsetAsync(agg, 0, ((size_t)N1 * DH + N1) * sizeof(float), stream);
        sage_scatter_add<<<E0, D_FEAT, 0, stream>>>(x, src0, dst0, E0, agg, deg);
        sage_gemm_wmma<256><<<(N1 + ROWS_PER_BLK - 1) / ROWS_PER_BLK, 256, 0, stream>>>(
            x, agg, deg, Ws0, Wn0, b0, h1, N1, 0);
    }

    // ---- Layer 1: h1 -> h2 (N2 x 256), ReLU fused ----
    {
        float* deg = agg + (size_t)N2 * DH;
        hipMemsetAsync(agg, 0, ((size_t)N2 * DH + N2) * sizeof(float), stream);
        sage_scatter_add<<<E1, D_FEAT, 0, stream>>>(h1, src1, dst1, E1, agg, deg);
        sage_gemm_wmma<256><<<(N2 + ROWS_PER_BLK - 1) / ROWS_PER_BLK, 256, 0, stream>>>(
            h1, agg, deg, Ws1, Wn1, b1, h2, N2, 1);
    }

    // ---- Layer 2: h2 -> out (N3 x 64) ----
    {
        float* deg = agg + (size_t)N3 * DH;
        hipMemsetAsync(agg, 0, ((size_t)N3 * DH + N3) * sizeof(float), stream);
        sage_scatter_add<<<E2, D_FEAT, 0, stream>>>(h2, src2, dst2, E2, agg, deg);
        sage_gemm_wmma<64><<<(N3 + ROWS_PER_BLK - 1) / ROWS_PER_BLK, 256, 0, stream>>>(
            h2, agg, deg, Ws2, Wn2, b2, out, N3, 0);
    }
}